// DGMRF_32581621907834
// MI455X (gfx1250) — compile-verified
//
#include <hip/hip_runtime.h>
#include <hip/hip_bf16.h>

// DGMRF graph propagation for MI455X (gfx1250, wave32).
//
// Design:
//  * deg/log_deg computed once (scatter count at src).
//  * x transposed to node-major xt[N][16] -> every per-edge access is a
//    contiguous 64B vector.
//  * Edge weight exp((dp-1)*log_deg[dst]) depends only on dst, so the
//    aggregation accumulates RAW neighbor sums S; weights fold into combine.
//  * Aggregation: build a dst-major CSR ONCE (indeg count -> LDS scan ->
//    atomic-position fill), then each layer is a pure gather (no float
//    atomics): 16 lanes per node (one per channel) stream the incoming src
//    list, broadcasting indices with 16-wide __shfl. ~205MB/layer of plain
//    L2 loads instead of 51.2M L2 atomic RMWs per layer.
//  * Combine per 16-node x 16-channel tile: Out = diag(c1)*X + diag(c2)*S + b
//    via chained V_WMMA_F32_16X16X4_F32, one wave per tile.
//  * Working set (~47MB incl. CSR) is L2-resident (192MB L2).
//  * If ws_size can't hold the CSR, fall back to the atomic-scatter path.

typedef float v2f __attribute__((ext_vector_type(2)));
typedef float v8f __attribute__((ext_vector_type(8)));

#define TCH 16

__global__ void k_zero_i32(int* __restrict__ p, int n) {
  int i = blockIdx.x * blockDim.x + threadIdx.x;
  if (i < n) p[i] = 0;
}

__global__ void k_count(const int* __restrict__ idx, int* __restrict__ cnt, int E) {
  int e = blockIdx.x * blockDim.x + threadIdx.x;
  if (e >= E) return;
  __builtin_prefetch(idx + e + 4096, 0, 0);
  atomicAdd(&cnt[idx[e]], 1);
}

// log_deg[n] = log(deg[n]); xt[n][t] = x[t][n]  (transpose to node-major)
__global__ void k_prep(const float* __restrict__ x, const int* __restrict__ deg,
                       float* __restrict__ log_deg, float* __restrict__ xt, int n) {
  int i = blockIdx.x * blockDim.x + threadIdx.x;
  if (i >= n) return;
  log_deg[i] = logf((float)deg[i]);
  float4* o = (float4*)(xt + (size_t)i * TCH);
#pragma unroll
  for (int q = 0; q < 4; ++q) {
    float4 v;
    v.x = x[(size_t)(4 * q + 0) * n + i];
    v.y = x[(size_t)(4 * q + 1) * n + i];
    v.z = x[(size_t)(4 * q + 2) * n + i];
    v.w = x[(size_t)(4 * q + 3) * n + i];
    o[q] = v;
  }
}

// Single-workgroup exclusive scan over indeg -> offsets[0..n], offsets[n]=E.
// 1024 threads, Hillis-Steele in LDS per 1024-chunk, serial carry.
__global__ void k_scan(const int* __restrict__ cnt, int* __restrict__ offsets, int n) {
  __shared__ int buf[1024];
  __shared__ int carry_s;
  const int tid = threadIdx.x;
  if (tid == 0) carry_s = 0;
  __syncthreads();
  for (int base = 0; base < n; base += 1024) {
    int i = base + tid;
    int v = (i < n) ? cnt[i] : 0;
    buf[tid] = v;
    __syncthreads();
#pragma unroll
    for (int off = 1; off < 1024; off <<= 1) {
      int add = (tid >= off) ? buf[tid - off] : 0;
      __syncthreads();
      buf[tid] += add;
      __syncthreads();
    }
    int incl = buf[tid];
    int c = carry_s;
    __syncthreads();
    if (i < n) offsets[i] = c + incl - v;  // exclusive prefix
    if (tid == 0) carry_s = c + buf[1023];
    __syncthreads();
  }
  if (tid == 0) offsets[n] = carry_s;
}

__global__ void k_copy_i32(const int* __restrict__ a, int* __restrict__ b, int n) {
  int i = blockIdx.x * blockDim.x + threadIdx.x;
  if (i < n) b[i] = a[i];
}

// Fill CSR: csr[slot] = src for each edge, slot = atomic position in dst's run.
__global__ void k_fill(const int* __restrict__ src, const int* __restrict__ dst,
                       int* __restrict__ cursor, int* __restrict__ csr, int E) {
  int e = blockIdx.x * blockDim.x + threadIdx.x;
  if (e >= E) return;
  __builtin_prefetch(src + e + 4096, 0, 0);
  __builtin_prefetch(dst + e + 4096, 0, 0);
  int d = dst[e];
  int pos = atomicAdd(&cursor[d], 1);
  csr[pos] = src[e];
}

// Gather aggregation: S[n][t] = sum over incoming srcs of xt[src][t].
// One half-wave (16 lanes = 16 channels) per node; indices loaded 16 at a
// time cooperatively and broadcast with 16-wide shfl (LDS-permute path).
__global__ void k_gather(const int* __restrict__ offsets, const int* __restrict__ csr,
                         const float* __restrict__ xt, float* __restrict__ S, int n) {
  int lane = threadIdx.x & 31;
  int wave = (blockIdx.x * blockDim.x + threadIdx.x) >> 5;
  int half = lane >> 4;
  int t = lane & 15;
  int node = wave * 2 + half;
  if (node >= n) return;
  int start = offsets[node];
  int end = offsets[node + 1];
  float acc = 0.0f;
  int j = start;
  // Full 16-index chunks: no bounds checks in the hot loop.
  for (; j + 16 <= end; j += 16) {
    int myidx = csr[j + t];
#pragma unroll
    for (int i = 0; i < 16; ++i) {
      int s = __shfl(myidx, i, 16);
      acc += xt[(size_t)s * TCH + t];
    }
  }
  if (j < end) {
    int jj = j + t;
    int myidx = (jj < end) ? csr[jj] : -1;
#pragma unroll
    for (int i = 0; i < 16; ++i) {
      int s = __shfl(myidx, i, 16);
      if (s >= 0) acc += xt[(size_t)s * TCH + t];  // uniform per 16-lane group
    }
  }
  S[(size_t)node * TCH + t] = acc;
}

// ---- Fallback path (used only if ws too small for CSR) ----
__global__ void k_zero4(float4* __restrict__ p, int n4) {
  int i = blockIdx.x * blockDim.x + threadIdx.x;
  if (i < n4) p[i] = make_float4(0.f, 0.f, 0.f, 0.f);
}

__global__ void k_scatter(const int* __restrict__ src, const int* __restrict__ dst,
                          const float* __restrict__ xt, float* __restrict__ S, int E) {
  int gid = blockIdx.x * blockDim.x + threadIdx.x;
  int e = gid >> 2;
  int q = gid & 3;
  if (e >= E) return;
  __builtin_prefetch(src + e + 4096, 0, 0);
  __builtin_prefetch(dst + e + 4096, 0, 0);
  int s = src[e];
  int d = dst[e];
  const float4 v = *(const float4*)(xt + (size_t)s * TCH + q * 4);
  float* p = S + (size_t)d * TCH + q * 4;
  atomicAdd(p + 0, v.x);
  atomicAdd(p + 1, v.y);
  atomicAdd(p + 2, v.z);
  atomicAdd(p + 3, v.w);
}

// Combine per 16x16 tile via WMMA:  Out = diag(c1)*X + diag(c2)*S + b
//   c1[n] = exp(a1) * exp(dp*ld[n]),  c2[n] = exp(a1)*tanh(a1) * exp((dp-1)*ld[n])
__global__ void k_combine(const float* __restrict__ xt, const float* __restrict__ S,
                          const float* __restrict__ log_deg,
                          const float* __restrict__ alpha1, const float* __restrict__ gamma,
                          const float* __restrict__ bias,
                          int layer, int n, float* __restrict__ out, int final_mode) {
  int lane = threadIdx.x & 31;
  int wave = (blockIdx.x * blockDim.x + threadIdx.x) >> 5;
  int ntiles = (n + 15) >> 4;
  if (wave >= ntiles) return;
  int base = wave << 4;

  float a1 = alpha1[layer];
  float g = gamma[layer];
  float bc = bias[layer];
  float dp = 1.0f / (1.0f + expf(-g));  // sigmoid(gamma)
  float sw = expf(a1);                  // self weight
  float nw = sw * tanhf(a1);            // neighbor weight (shared-alpha quirk kept)

  int m = lane & 15;
  int half = lane >> 4;

  if (base + 16 <= n) {
    float ld = log_deg[base + m];
    float c1 = sw * expf(dp * ld);
    float c2 = nw * expf((dp - 1.0f) * ld);

    v8f acc = {0.f, 0.f, 0.f, 0.f, 0.f, 0.f, 0.f, 0.f};

    // A 16x4: lanes0-15 hold K={k0,k0+1}; lanes16-31 hold K={k0+2,k0+3}.
    // B 4x16 (symmetric): v[0]=row k0+2*half, v[1]=row k0+1+2*half, col=lane&15.
#pragma unroll
    for (int k0 = 0; k0 < 16; k0 += 4) {
      int r0 = k0 + 2 * half;
      v2f a, b;
      a.x = (r0 == m) ? c1 : 0.0f;
      a.y = (r0 + 1 == m) ? c1 : 0.0f;
      b.x = xt[(size_t)(base + r0) * TCH + m];
      b.y = xt[(size_t)(base + r0 + 1) * TCH + m];
      acc = __builtin_amdgcn_wmma_f32_16x16x4_f32(false, a, false, b, (short)0, acc,
                                                  false, false);
    }
#pragma unroll
    for (int k0 = 0; k0 < 16; k0 += 4) {
      int r0 = k0 + 2 * half;
      v2f a, b;
      a.x = (r0 == m) ? c2 : 0.0f;
      a.y = (r0 + 1 == m) ? c2 : 0.0f;
      b.x = S[(size_t)(base + r0) * TCH + m];
      b.y = S[(size_t)(base + r0 + 1) * TCH + m];
      acc = __builtin_amdgcn_wmma_f32_16x16x4_f32(false, a, false, b, (short)0, acc,
                                                  false, false);
    }

    // C/D: VGPR j, lanes0-15 -> M=j, lanes16-31 -> M=j+8; col = lane&15.
#pragma unroll
    for (int j = 0; j < 8; ++j) {
      int node = base + j + 8 * half;
      float v = acc[j] + bc;
      if (final_mode) out[(size_t)m * n + node] = v;    // back to [T, N]
      else            out[(size_t)node * TCH + m] = v;  // node-major
    }
  } else {
    if (half == 0) {  // ragged tail, scalar per-channel
      int t = m;
      for (int node = base; node < n; ++node) {
        float ld = log_deg[node];
        float c1 = sw * expf(dp * ld);
        float c2 = nw * expf((dp - 1.0f) * ld);
        float v = c1 * xt[(size_t)node * TCH + t] + c2 * S[(size_t)node * TCH + t] + bc;
        if (final_mode) out[(size_t)t * n + node] = v;
        else            out[(size_t)node * TCH + t] = v;
      }
    }
  }
}

static inline size_t a16(size_t x) { return (x + 15) & ~(size_t)15; }

extern "C" void kernel_launch(void* const* d_in, const int* in_sizes, int n_in,
                              void* d_out, int out_size, void* d_ws, size_t ws_size,
                              hipStream_t stream) {
  (void)n_in; (void)out_size;

  const int N = in_sizes[0] / TCH;  // nodes
  const int E = in_sizes[1] / 2;    // edges
  const int L = in_sizes[2];        // layers

  const float* x = (const float*)d_in[0];
  const int* edge = (const int*)d_in[1];
  const int* src = edge;
  const int* dst = edge + E;
  const float* alpha1 = (const float*)d_in[2];
  const float* gamma = (const float*)d_in[3];
  const float* bias = (const float*)d_in[4];

  char* ws = (char*)d_ws;
  const int B = 256;

  // CSR-path workspace layout.
  size_t o_deg = 0;                               // int[N]  out-degree (for log_deg)
  size_t o_indeg = a16(o_deg + 4ull * N);         // int[N]  in-degree (CSR sizing)
  size_t o_off = a16(o_indeg + 4ull * N);         // int[N+1] CSR offsets
  size_t o_cur = a16(o_off + 4ull * (N + 1));     // int[N]  fill cursors
  size_t o_csr = a16(o_cur + 4ull * N);           // int[E]  src lists by dst
  size_t o_ld = a16(o_csr + 4ull * E);            // float[N] log_deg
  size_t o_xt = a16(o_ld + 4ull * N);             // float[N*16]
  size_t o_S = a16(o_xt + 64ull * N);             // float[N*16]
  size_t need_csr = o_S + 64ull * N;
  const bool use_csr = (ws_size >= need_csr);

  if (use_csr) {
    int* deg = (int*)(ws + o_deg);
    int* indeg = (int*)(ws + o_indeg);
    int* offsets = (int*)(ws + o_off);
    int* cursor = (int*)(ws + o_cur);
    int* csr = (int*)(ws + o_csr);
    float* log_deg = (float*)(ws + o_ld);
    float* xt = (float*)(ws + o_xt);
    float* S = (float*)(ws + o_S);

    k_zero_i32<<<(N + B - 1) / B, B, 0, stream>>>(deg, N);
    k_zero_i32<<<(N + B - 1) / B, B, 0, stream>>>(indeg, N);
    k_count<<<(E + B - 1) / B, B, 0, stream>>>(src, deg, E);
    k_count<<<(E + B - 1) / B, B, 0, stream>>>(dst, indeg, E);
    k_prep<<<(N + B - 1) / B, B, 0, stream>>>(x, deg, log_deg, xt, N);
    k_scan<<<1, 1024, 0, stream>>>(indeg, offsets, N);
    k_copy_i32<<<(N + B - 1) / B, B, 0, stream>>>(offsets, cursor, N);
    k_fill<<<(E + B - 1) / B, B, 0, stream>>>(src, dst, cursor, csr, E);

    for (int l = 0; l < L; ++l) {
      const int gwaves = (N + 1) / 2;  // one half-wave per node
      const int gthreads = gwaves * 32;
      k_gather<<<(gthreads + B - 1) / B, B, 0, stream>>>(offsets, csr, xt, S, N);

      const int ntiles = (N + 15) / 16;
      const int cthreads = ntiles * 32;
      const int last = (l == L - 1) ? 1 : 0;
      float* out = last ? (float*)d_out : xt;
      k_combine<<<(cthreads + B - 1) / B, B, 0, stream>>>(xt, S, log_deg, alpha1, gamma,
                                                          bias, l, N, out, last);
    }
  } else {
    // Fallback: atomic-scatter path (smaller workspace: deg | log_deg | xt | S).
    int* deg = (int*)ws;
    float* log_deg = (float*)(ws + 4ull * N);
    float* xt = (float*)(ws + 8ull * N);
    float* S = (float*)(ws + 8ull * N + 64ull * N);

    k_zero_i32<<<(N + B - 1) / B, B, 0, stream>>>(deg, N);
    k_count<<<(E + B - 1) / B, B, 0, stream>>>(src, deg, E);
    k_prep<<<(N + B - 1) / B, B, 0, stream>>>(x, deg, log_deg, xt, N);

    for (int l = 0; l < L; ++l) {
      const int n4 = N * 4;
      k_zero4<<<(n4 + B - 1) / B, B, 0, stream>>>((float4*)S, n4);
      const int nscatter = E * 4;
      k_scatter<<<(nscatter + B - 1) / B, B, 0, stream>>>(src, dst, xt, S, E);

      const int ntiles = (N + 15) / 16;
      const int cthreads = ntiles * 32;
      const int last = (l == L - 1) ? 1 : 0;
      float* out = last ? (float*)d_out : xt;
      k_combine<<<(cthreads + B - 1) / B, B, 0, stream>>>(xt, S, log_deg, alpha1, gamma,
                                                          bias, l, N, out, last);
    }
  }
}